// LearnedGateMemory_35270271435231
// MI455X (gfx1250) — compile-verified
//
#include <hip/hip_runtime.h>
#include <hip/hip_bf16.h>

#define BATCH 128
#define TLEN 4096
#define HID 512
#define MSLOTS 64
#define KWR 5
#define VOCAB 64

typedef float v2f __attribute__((ext_vector_type(2)));
typedef float v8f __attribute__((ext_vector_type(8)));

// ---------------------------------------------------------------------------
// Kernel 1: gate_probs = sigmoid(enc_hidden @ Wg + bg)   [B,T]
// One wave32 per token; each lane reads 4 float4 (16 floats) of the 512-row.
// Streams the 1 GiB enc_hidden tensor — the runtime-dominant kernel.
// ---------------------------------------------------------------------------
__global__ void gate_kernel(const float* __restrict__ enc,
                            const float* __restrict__ Wg,
                            const float* __restrict__ bg,
                            float* __restrict__ gate) {
    const int wave = (int)((blockIdx.x * blockDim.x + threadIdx.x) >> 5);
    const int lane = threadIdx.x & 31;
    if (wave >= BATCH * TLEN) return;
    const float4* row = (const float4*)(enc + (size_t)wave * HID);
    const float4* wg4 = (const float4*)Wg;
    float s = 0.f;
#pragma unroll
    for (int i = 0; i < 4; ++i) {
        const float4 v = row[lane + 32 * i];
        const float4 w = wg4[lane + 32 * i];
        s += v.x * w.x + v.y * w.y + v.z * w.z + v.w * w.w;
    }
#pragma unroll
    for (int off = 16; off >= 1; off >>= 1) s += __shfl_down(s, off, 32);
    if (lane == 0) gate[wave] = 1.f / (1.f + expf(-(s + bg[0])));
}

// ---------------------------------------------------------------------------
// Kernel 2: top-5 indices per batch row (iterative argmax, lowest index wins
// ties — matches jax.lax.top_k ordering). One 256-thread block per batch.
// ---------------------------------------------------------------------------
__global__ void topk_kernel(const float* __restrict__ gate,
                            int* __restrict__ top_idx) {
    const int b = blockIdx.x;
    const float* g = gate + (size_t)b * TLEN;
    __shared__ float sval[256];
    __shared__ int   sidx[256];
    __shared__ int   chosen[KWR];

    for (int kp = 0; kp < KWR; ++kp) {
        float best = -1e30f;
        int   bi   = 0x7fffffff;
        for (int t = threadIdx.x; t < TLEN; t += 256) {
            bool skip = false;
            for (int j = 0; j < kp; ++j) if (chosen[j] == t) skip = true;
            if (skip) continue;
            const float v = g[t];
            if (v > best || (v == best && t < bi)) { best = v; bi = t; }
        }
        sval[threadIdx.x] = best;
        sidx[threadIdx.x] = bi;
        __syncthreads();
        for (int s = 128; s > 0; s >>= 1) {
            if (threadIdx.x < s) {
                const float ov = sval[threadIdx.x + s];
                const int   oi = sidx[threadIdx.x + s];
                if (ov > sval[threadIdx.x] ||
                    (ov == sval[threadIdx.x] && oi < sidx[threadIdx.x])) {
                    sval[threadIdx.x] = ov;
                    sidx[threadIdx.x] = oi;
                }
            }
            __syncthreads();
        }
        if (threadIdx.x == 0) chosen[kp] = sidx[0];
        __syncthreads();
    }
    if (threadIdx.x < KWR) top_idx[b * KWR + threadIdx.x] = chosen[threadIdx.x];
}

// ---------------------------------------------------------------------------
// Kernel 3: memory[b,m,:] = enc[b, top_idx[b,m], :] for m<5, else 0.
// One 128-thread block per (b,m); one float4 per thread.
// ---------------------------------------------------------------------------
__global__ void memory_kernel(const float* __restrict__ enc,
                              const int* __restrict__ top_idx,
                              float* __restrict__ memory) {
    const int bm = blockIdx.x;
    const int b = bm / MSLOTS, m = bm % MSLOTS;
    float4* dst = (float4*)(memory + (size_t)bm * HID);
    if (m < KWR) {
        const int t = top_idx[b * KWR + m];
        const float4* src = (const float4*)(enc + ((size_t)b * TLEN + t) * HID);
        dst[threadIdx.x] = src[threadIdx.x];
    } else {
        dst[threadIdx.x] = make_float4(0.f, 0.f, 0.f, 0.f);
    }
}

// ---------------------------------------------------------------------------
// WMMA fp32 GEMM: C[M,N] = A[M,K] @ B[K,N] (+bias), optional B-transposed
// (TRANSB=1: Beff[k][n] = Bm[n*K + k]).  One wave per 16x16 C tile, K stepped
// by 4 via V_WMMA_F32_16X16X4_F32.  Fragment layouts per CDNA5 ISA 7.12.2:
//   A 16x4: lane<16 -> K=0,1 ; lane>=16 -> K=2,3 (2 VGPRs)
//   B 4x16 / C 16x16: rows striped across lanes within a VGPR.
// ---------------------------------------------------------------------------
template <int TRANSB>
__global__ void wmma_gemm_f32(const float* __restrict__ A,
                              const float* __restrict__ Bm,
                              const float* __restrict__ bias,
                              float* __restrict__ C,
                              int Mdim, int Ndim, int Kdim) {
    const int tiles_n = Ndim >> 4;
    const int tiles   = (Mdim >> 4) * tiles_n;
    const int wv = (int)((blockIdx.x * blockDim.x + threadIdx.x) >> 5);
    if (wv >= tiles) return;
    const int mt = wv / tiles_n, nt = wv % tiles_n;
    const int lane = threadIdx.x & 31;
    const int half = lane >> 4, idx = lane & 15;

    const int ar = mt * 16 + idx;   // A row this lane carries
    const int bn = nt * 16 + idx;   // B col this lane carries
    v8f acc = {0.f, 0.f, 0.f, 0.f, 0.f, 0.f, 0.f, 0.f};

    for (int k0 = 0; k0 < Kdim; k0 += 4) {
        const int kk = k0 + half * 2;
        v2f a, b;
        a.x = A[(size_t)ar * Kdim + kk];
        a.y = A[(size_t)ar * Kdim + kk + 1];
        if (TRANSB) {
            b.x = Bm[(size_t)bn * Kdim + kk];
            b.y = Bm[(size_t)bn * Kdim + kk + 1];
        } else {
            b.x = Bm[(size_t)kk * Ndim + bn];
            b.y = Bm[(size_t)(kk + 1) * Ndim + bn];
        }
        acc = __builtin_amdgcn_wmma_f32_16x16x4_f32(
            false, a, false, b, (short)0, acc, false, false);
    }
#pragma unroll
    for (int r = 0; r < 8; ++r) {
        const int row = mt * 16 + half * 8 + r;
        const int col = nt * 16 + idx;
        float v = acc[r];
        if (bias) v += bias[col];
        C[(size_t)row * Ndim + col] = v;
    }
}

// ---------------------------------------------------------------------------
// Kernel 6: scores (5 live + 1 shared empty-slot score), softmax over 64,
// retrieved = sum attn[m]*memory[b,m,:], out = retrieved + query_hidden.
// score[b,m] = (memory[b,m]·r[b] + q[b]·bk)/sqrt(H); empty slots: q·bk/sqrt(H).
// One 256-thread block per batch.
// ---------------------------------------------------------------------------
__global__ void attend_kernel(const float* __restrict__ q,
                              const float* __restrict__ r,
                              const float* __restrict__ bk,
                              const float* __restrict__ memory,
                              const float* __restrict__ query_hidden,
                              float* __restrict__ out512) {
    const int b = blockIdx.x;
    __shared__ float red[256];
    __shared__ float scores[KWR + 1];
    __shared__ float attn[KWR];
    const float inv_sqrt_h = 0.044194173824159216f;  // 1/sqrt(512)

    // empty-slot score: q . bk / sqrt(H)
    {
        float s = 0.f;
        for (int h = threadIdx.x; h < HID; h += 256)
            s += q[(size_t)b * HID + h] * bk[h];
        red[threadIdx.x] = s; __syncthreads();
        for (int st = 128; st > 0; st >>= 1) {
            if (threadIdx.x < st) red[threadIdx.x] += red[threadIdx.x + st];
            __syncthreads();
        }
        if (threadIdx.x == 0) scores[KWR] = red[0] * inv_sqrt_h;
        __syncthreads();
    }
    // live-slot scores
    for (int m = 0; m < KWR; ++m) {
        const float* g = memory + ((size_t)b * MSLOTS + m) * HID;
        float s = 0.f;
        for (int h = threadIdx.x; h < HID; h += 256)
            s += g[h] * r[(size_t)b * HID + h];
        red[threadIdx.x] = s; __syncthreads();
        for (int st = 128; st > 0; st >>= 1) {
            if (threadIdx.x < st) red[threadIdx.x] += red[threadIdx.x + st];
            __syncthreads();
        }
        if (threadIdx.x == 0) scores[m] = red[0] * inv_sqrt_h + scores[KWR];
        __syncthreads();
    }
    // softmax over 64 = {5 live, 59 copies of scores[KWR]}
    if (threadIdx.x == 0) {
        float mx = scores[KWR];
        for (int m = 0; m < KWR; ++m) mx = fmaxf(mx, scores[m]);
        float denom = (float)(MSLOTS - KWR) * expf(scores[KWR] - mx);
        float e[KWR];
        for (int m = 0; m < KWR; ++m) { e[m] = expf(scores[m] - mx); denom += e[m]; }
        for (int m = 0; m < KWR; ++m) attn[m] = e[m] / denom;
    }
    __syncthreads();
    // retrieved + query  (empty slots contribute zero vectors)
    for (int h = threadIdx.x; h < HID; h += 256) {
        float acc = 0.f;
        for (int m = 0; m < KWR; ++m)
            acc += attn[m] * memory[((size_t)b * MSLOTS + m) * HID + h];
        out512[(size_t)b * HID + h] = acc + query_hidden[(size_t)b * HID + h];
    }
}

// ---------------------------------------------------------------------------
// Launch
// ---------------------------------------------------------------------------
extern "C" void kernel_launch(void* const* d_in, const int* in_sizes, int n_in,
                              void* d_out, int out_size, void* d_ws, size_t ws_size,
                              hipStream_t stream) {
    const float* enc   = (const float*)d_in[0];  // [B,T,H]
    const float* query = (const float*)d_in[1];  // [B,H]
    const float* Wg    = (const float*)d_in[2];  // [H,1]
    const float* bg    = (const float*)d_in[3];  // [1]
    const float* Wq    = (const float*)d_in[4];  // [H,H]
    const float* bq    = (const float*)d_in[5];  // [H]
    const float* Wk    = (const float*)d_in[6];  // [H,H]
    const float* bk    = (const float*)d_in[7];  // [H]
    const float* Wo    = (const float*)d_in[8];  // [H,V]
    const float* bo    = (const float*)d_in[9];  // [V]

    // d_out layout: logits [B*V] | gate_probs [B*T] | memory [B*M*H]
    float* logits = (float*)d_out;
    float* gate   = logits + (size_t)BATCH * VOCAB;
    float* memory = gate + (size_t)BATCH * TLEN;

    // workspace layout
    float* q_ws   = (float*)d_ws;                       // [B*H]
    float* r_ws   = q_ws + (size_t)BATCH * HID;         // [B*H]
    float* out512 = r_ws + (size_t)BATCH * HID;         // [B*H]
    int*   topidx = (int*)(out512 + (size_t)BATCH * HID); // [B*KWR]

    // 1) gate probs: B*T waves, 8 waves per 256-thread block
    gate_kernel<<<(BATCH * TLEN) / 8, 256, 0, stream>>>(enc, Wg, bg, gate);

    // 2) top-5 per batch
    topk_kernel<<<BATCH, 256, 0, stream>>>(gate, topidx);

    // 3) build memory tensor (gather + zero fill)
    memory_kernel<<<BATCH * MSLOTS, 128, 0, stream>>>(enc, topidx, memory);

    // 4) q = query @ Wq + bq      (128x512x512) -> 256 tiles, 4 waves/block
    wmma_gemm_f32<0><<<64, 128, 0, stream>>>(query, Wq, bq, q_ws,
                                             BATCH, HID, HID);

    // 5) r = q @ Wk^T             (128x512x512, transposed B, no bias)
    wmma_gemm_f32<1><<<64, 128, 0, stream>>>(q_ws, Wk, nullptr, r_ws,
                                             BATCH, HID, HID);

    // 6) scores + softmax + retrieve + residual
    attend_kernel<<<BATCH, 256, 0, stream>>>(q_ws, r_ws, bk, memory, query,
                                             out512);

    // 7) logits = out512 @ Wo + bo  (128x64x512) -> 32 tiles, 4 waves/block
    wmma_gemm_f32<0><<<8, 128, 0, stream>>>(out512, Wo, bo, logits,
                                            BATCH, VOCAB, HID);
}